// GSN_14783277433402
// MI455X (gfx1250) — compile-verified
//
#include <hip/hip_runtime.h>
#include <hip/hip_bf16.h>

typedef __attribute__((ext_vector_type(2))) float v2f;
typedef __attribute__((ext_vector_type(8))) float v8f;

#define NN 50000
#define EE 800000
#define GG 64
#define HH 128

// ---------------------------------------------------------------- degree + edge-feature scatter
__global__ void k_deg_xa(const int* __restrict__ src, const float* __restrict__ edge_attr,
                         float* __restrict__ deg, float* __restrict__ xa) {
    int e = blockIdx.x * blockDim.x + threadIdx.x;
    if (e >= EE) return;
    int s = src[e];
    atomicAdd(&deg[s], 1.0f);
#pragma unroll
    for (int f = 0; f < 4; ++f)
        atomicAdd(&xa[s * 4 + f], edge_attr[e * 4 + f]);
}

__global__ void k_norm(const float* __restrict__ deg, float* __restrict__ norm) {
    int i = blockIdx.x * blockDim.x + threadIdx.x;
    if (i >= NN) return;
    float d = deg[i];
    norm[i] = (d > 0.0f) ? rsqrtf(d) : 0.0f;
}

__global__ void k_norm_e(const int* __restrict__ src, const int* __restrict__ dst,
                         const float* __restrict__ norm, float* __restrict__ norm_e) {
    int e = blockIdx.x * blockDim.x + threadIdx.x;
    if (e >= EE) return;
    norm_e[e] = norm[src[e]] * norm[dst[e]];
}

// ---------------------------------------------------------------- edge scatter, 9-wide (layer 0 input)
__global__ void k_scatter9(const int* __restrict__ src, const int* __restrict__ dst,
                           const float* __restrict__ ne, const float* __restrict__ x,
                           float* __restrict__ agg) {
    int e = blockIdx.x * blockDim.x + threadIdx.x;
    if (e >= EE) return;
    int s = src[e], d = dst[e];
    float w = ne[e];
#pragma unroll
    for (int f = 0; f < 9; ++f)
        atomicAdd(&agg[d * 9 + f], w * x[s * 9 + f]);
}

// ---------------------------------------------------------------- edge scatter, 128-wide (layer 1). 2 edges / 256-thr block
__global__ void k_scatter128(const int* __restrict__ src, const int* __restrict__ dst,
                             const float* __restrict__ ne, const float* __restrict__ h,
                             float* __restrict__ agg) {
    int e = blockIdx.x * 2 + (threadIdx.x >> 7);
    int f = threadIdx.x & 127;
    if (e >= EE) return;
    int s = src[e], d = dst[e];
    atomicAdd(&agg[d * HH + f], ne[e] * h[s * HH + f]);
}

// ---------------------------------------------------------------- pack layer-0 A:  A0[N,24] = [x(9) | agg_x(9) | xa(4) | 0 0]
__global__ void k_pack_a0(const float* __restrict__ x, const float* __restrict__ agg_x,
                          const float* __restrict__ xa, float* __restrict__ A0) {
    int idx = blockIdx.x * blockDim.x + threadIdx.x;   // N*24
    if (idx >= NN * 24) return;
    int i = idx / 24, c = idx - i * 24;
    float v = 0.0f;
    if (c < 9)       v = x[i * 9 + c];
    else if (c < 18) v = agg_x[i * 9 + (c - 9)];
    else if (c < 22) v = xa[i * 4 + (c - 18)];
    A0[idx] = v;
}

// ---------------------------------------------------------------- weight pre-combination, pair-interleaved layout:
//   Wp[(k>>1)*256 + h*2 + (k&1)] = W[k][h]   -> lane's B fragment is one contiguous 8-byte load
// Wcat0 [24,128]: rows 0-8 = w0[0]+w0[1]-w0[2]; 9-17 = 2*w0[2]; 18-21 = sum_k ew0[k]; 22-23 = 0
__global__ void k_prep_w0(const float* __restrict__ w0, const float* __restrict__ ew0,
                          float* __restrict__ Wp) {
    int idx = blockIdx.x * blockDim.x + threadIdx.x;
    if (idx >= 24 * HH) return;
    int k = idx >> 7, h = idx & 127;
    float v = 0.0f;
    if (k < 9)
        v = w0[k * HH + h] + w0[9 * HH + k * HH + h] - w0[18 * HH + k * HH + h];
    else if (k < 18)
        v = 2.0f * w0[18 * HH + (k - 9) * HH + h];
    else if (k < 22) {
        int kk = k - 18;
        v = ew0[kk * HH + h] + ew0[4 * HH + kk * HH + h] + ew0[8 * HH + kk * HH + h];
    }
    Wp[(k >> 1) * 256 + h * 2 + (k & 1)] = v;
}

// Wcat1 [260,128]: rows 0-127 = w1[0]+w1[1]-w1[2]; 128-255 = 2*w1[2]; 256-259 = sum_k ew1[k]
__global__ void k_prep_w1(const float* __restrict__ w1, const float* __restrict__ ew1,
                          float* __restrict__ Wp) {
    int idx = blockIdx.x * blockDim.x + threadIdx.x;
    if (idx >= 260 * HH) return;
    int k = idx >> 7, h = idx & 127;
    float v;
    if (k < 128)
        v = w1[k * HH + h] + w1[128 * HH + k * HH + h] - w1[256 * HH + k * HH + h];
    else if (k < 256)
        v = 2.0f * w1[256 * HH + (k - 128) * HH + h];
    else {
        int kk = k - 256;
        v = ew1[kk * HH + h] + ew1[4 * HH + kk * HH + h] + ew1[8 * HH + kk * HH + h];
    }
    Wp[(k >> 1) * 256 + h * 2 + (k & 1)] = v;
}

// ---------------------------------------------------------------- fused multi-section GEMM via V_WMMA_F32_16X16X4_F32
// one section:  acc += A[16 rows, KC] @ W[KC, 128], branch-free, all v2f (b64) loads
__device__ __forceinline__ void gemm_section(const float* __restrict__ A, int lda,
                                             const float* __restrict__ Wp,  // pair-packed, section base
                                             int KC, int m, int n, int ks, v8f acc[8]) {
    const float* arow = A + (size_t)m * lda + ks;
    for (int kk = 0; kk < KC; kk += 4) {
        v2f a = *(const v2f*)(arow + kk);                        // A[m][kk+ks .. kk+ks+1]
        const float* wrow = Wp + (size_t)((kk + ks) >> 1) * 256; // pair row, 128 cols x 2
#pragma unroll
        for (int t = 0; t < 8; ++t) {
            v2f b = *(const v2f*)(wrow + (t * 16 + n) * 2);
            acc[t] = __builtin_amdgcn_wmma_f32_16x16x4_f32(
                false, a, false, b, (short)0, acc[t], false, false);
        }
    }
}

// one wave computes a 16-row x 128-col strip of C = relu( [A1|A2|A3] @ W + bias )
template <int K1, int K2, int K3>
__global__ __launch_bounds__(256) void k_gemm_fused(
        const float* __restrict__ A1, const float* __restrict__ A2,
        const float* __restrict__ A3, const float* __restrict__ Wp,
        const float* __restrict__ bias, float* __restrict__ out, int nrows) {
    const int lane  = threadIdx.x & 31;
    const int wave  = threadIdx.x >> 5;
    const int strip = blockIdx.x * (blockDim.x >> 5) + wave;
    const int row0  = strip * 16;
    if (row0 >= nrows) return;               // wave-uniform (strip uniform per wave)

    const int m  = row0 + (lane & 15);       // A row this lane supplies
    const int n  = lane & 15;                // C column within tile
    const int ks = (lane >> 4) << 1;         // lanes 0-15 hold K{0,1}, lanes 16-31 hold K{2,3}

    const v8f vzero = {0.f, 0.f, 0.f, 0.f, 0.f, 0.f, 0.f, 0.f};
    v8f acc[8];
#pragma unroll
    for (int t = 0; t < 8; ++t) acc[t] = vzero;

    gemm_section(A1, K1, Wp, K1, m, n, ks, acc);
    if (K2 > 0) gemm_section(A2, K2, Wp + (K1 >> 1) * 256, K2, m, n, ks, acc);
    if (K3 > 0) gemm_section(A3, K3, Wp + ((K1 + K2) >> 1) * 256, K3, m, n, ks, acc);

    const int rbase = (lane >> 4) ? 8 : 0;   // C/D layout: VGPR r -> rows r / r+8 by lane half
#pragma unroll
    for (int t = 0; t < 8; ++t) {
        const int col = t * 16 + n;
        const float bb = bias[col];
#pragma unroll
        for (int r = 0; r < 8; ++r) {
            const int row = row0 + rbase + r;
            if (row < nrows) {
                float v = fmaxf(acc[t][r] + bb, 0.0f);
                out[(size_t)row * HH + col] = v;
            }
        }
    }
}

// ---------------------------------------------------------------- mean pool by graph
__global__ void k_pool(const float* __restrict__ h, const int* __restrict__ batch,
                       float* __restrict__ pool, float* __restrict__ counts) {
    int i = blockIdx.x;          // node
    int t = threadIdx.x;         // feature (blockDim = 128)
    if (i >= NN) return;
    int b = batch[i];
    atomicAdd(&pool[b * HH + t], h[(size_t)i * HH + t]);
    if (t == 0) atomicAdd(&counts[b], 1.0f);
}

// ---------------------------------------------------------------- final linear + log_softmax (G=64, C=4)
__global__ void k_final(const float* __restrict__ pool, const float* __restrict__ counts,
                        const float* __restrict__ lin_w, const float* __restrict__ lin_b,
                        float* __restrict__ out) {
    int g = threadIdx.x;
    if (g >= GG) return;
    float inv = 1.0f / fmaxf(counts[g], 1.0f);
    float logit[4];
#pragma unroll
    for (int c = 0; c < 4; ++c) logit[c] = lin_b[c];
    for (int hh = 0; hh < HH; ++hh) {
        float p = pool[g * HH + hh] * inv;
#pragma unroll
        for (int c = 0; c < 4; ++c) logit[c] += p * lin_w[hh * 4 + c];
    }
    float mx = logit[0];
#pragma unroll
    for (int c = 1; c < 4; ++c) mx = fmaxf(mx, logit[c]);
    float s = 0.0f;
#pragma unroll
    for (int c = 0; c < 4; ++c) s += expf(logit[c] - mx);
    float lse = logf(s);
#pragma unroll
    for (int c = 0; c < 4; ++c) out[g * 4 + c] = logit[c] - mx - lse;
}

// ================================================================ host
extern "C" void kernel_launch(void* const* d_in, const int* in_sizes, int n_in,
                              void* d_out, int out_size, void* d_ws, size_t ws_size,
                              hipStream_t stream) {
    const float* x     = (const float*)d_in[0];
    const float* eattr = (const float*)d_in[1];
    const float* w0    = (const float*)d_in[2];
    const float* ew0   = (const float*)d_in[3];
    const float* b0    = (const float*)d_in[4];
    const float* w1    = (const float*)d_in[5];
    const float* ew1   = (const float*)d_in[6];
    const float* b1    = (const float*)d_in[7];
    const float* lin_w = (const float*)d_in[8];
    const float* lin_b = (const float*)d_in[9];
    const int*   eidx  = (const int*)d_in[10];
    const int*   batch = (const int*)d_in[11];
    const int* src = eidx;
    const int* dst = eidx + EE;

    // workspace layout (floats); every segment is an even float count -> 8B alignment preserved
    float* ws = (float*)d_ws;
    size_t off = 0;
    float* deg    = ws + off; off += NN;
    float* norm   = ws + off; off += NN;
    float* norm_e = ws + off; off += EE;
    float* xa     = ws + off; off += (size_t)NN * 4;
    float* agg_x  = ws + off; off += (size_t)NN * 10;   // 9 used, padded even
    float* A0     = ws + off; off += (size_t)NN * 24;
    float* h0     = ws + off; off += (size_t)NN * HH;
    float* agg_h  = ws + off; off += (size_t)NN * HH;
    float* h1     = ws + off; off += (size_t)NN * HH;
    float* Wp0    = ws + off; off += 24 * HH;
    float* Wp1    = ws + off; off += 260 * HH;
    float* pool   = ws + off; off += GG * HH;
    float* counts = ws + off; off += GG;

    // zero accumulation buffers (capture-legal async memsets)
    hipMemsetAsync(deg,   0, NN * sizeof(float), stream);
    hipMemsetAsync(xa,    0, (size_t)NN * 4 * sizeof(float), stream);
    hipMemsetAsync(agg_x, 0, (size_t)NN * 9 * sizeof(float), stream);
    hipMemsetAsync(agg_h, 0, (size_t)NN * HH * sizeof(float), stream);
    hipMemsetAsync(pool,  0, (GG * HH + GG) * sizeof(float), stream);  // pool + counts contiguous

    // degree + per-src edge-feature aggregation (shared by both layers)
    k_deg_xa<<<(EE + 255) / 256, 256, 0, stream>>>(src, eattr, deg, xa);
    k_norm<<<(NN + 255) / 256, 256, 0, stream>>>(deg, norm);
    k_norm_e<<<(EE + 255) / 256, 256, 0, stream>>>(src, dst, norm, norm_e);

    // pre-combined, pair-interleaved weights
    k_prep_w0<<<(24 * HH + 255) / 256, 256, 0, stream>>>(w0, ew0, Wp0);
    k_prep_w1<<<(260 * HH + 255) / 256, 256, 0, stream>>>(w1, ew1, Wp1);

    const int strips = (NN + 15) / 16;       // 3125
    const int ggrid  = (strips + 7) / 8;     // 8 waves per 256-thr block

    // ---- layer 0
    k_scatter9<<<(EE + 255) / 256, 256, 0, stream>>>(src, dst, norm_e, x, agg_x);
    k_pack_a0<<<(NN * 24 + 255) / 256, 256, 0, stream>>>(x, agg_x, xa, A0);
    k_gemm_fused<24, 0, 0><<<ggrid, 256, 0, stream>>>(A0, nullptr, nullptr, Wp0, b0, h0, NN);

    // ---- layer 1
    k_scatter128<<<(EE + 1) / 2, 256, 0, stream>>>(src, dst, norm_e, h0, agg_h);
    k_gemm_fused<128, 128, 4><<<ggrid, 256, 0, stream>>>(h0, agg_h, xa, Wp1, b1, h1, NN);

    // ---- pool + head
    k_pool<<<NN, HH, 0, stream>>>(h1, batch, pool, counts);
    k_final<<<1, GG, 0, stream>>>(pool, counts, lin_w, lin_b, (float*)d_out);
}